// inv_comp_9543417332552
// MI455X (gfx1250) — compile-verified
//
#include <hip/hip_runtime.h>
#include <hip/hip_bf16.h>

typedef float v2f __attribute__((ext_vector_type(2)));
typedef float v8f __attribute__((ext_vector_type(8)));

#define LOG2E 1.4426950408889634f
#define LN2   0.6931471805599453f

// Numerically stable softplus using raw HW transcendentals:
//   softplus(x) = max(x,0) + ln(1 + exp(-|x|))
// The log argument is in [1,2] -> no denorm/negative guards needed, so we can
// use v_exp_f32 / v_log_f32 directly instead of the guarded OCML expansion.
__device__ __forceinline__ float splus(float x) {
    float ax = __builtin_fabsf(x);
    float e  = __builtin_amdgcn_exp2f(ax * -LOG2E);   // exp(-|x|) in (0,1]
    float l  = __builtin_amdgcn_logf(1.f + e);        // log2(1+e), arg in [1,2]
    return __builtin_fmaxf(x, 0.f) + LN2 * l;
}

__device__ __forceinline__ float sigmoidf(float x) {
    return 1.f / (1.f + __builtin_amdgcn_exp2f(-x * LOG2E));
}

// lane <-> lane^16 exchange (group-of-32 swizzle, SWAPX16: xor=0x10,and=0x1f)
__device__ __forceinline__ float swap16(float x) {
    int i = __builtin_amdgcn_ds_swizzle(__float_as_int(x), 0x401f);
    return __int_as_float(i);
}

// --------------------------------------------------------------------------
// One-thread setup kernel: compute the two batch-invariant constants
//   cst[0] = e_id  (MLP energy at F = I, i.e. x = [3, 1, -1, 3])
//   cst[1] = h_c   (H = -grad e_id = h_c * I; h_c = -(2g0 + g1 - g2 + 4g3))
// --------------------------------------------------------------------------
__global__ void hyper_setup_kernel(const float* __restrict__ W0, const float* __restrict__ b0,
                                   const float* __restrict__ Wh, const float* __restrict__ bh,
                                   const float* __restrict__ Wf, const float* __restrict__ bf,
                                   float* __restrict__ cst) {
    if (threadIdx.x != 0 || blockIdx.x != 0) return;
    const float x[4] = {3.f, 1.f, -1.f, 3.f};

    float pre0[5], h[5];
    for (int n = 0; n < 5; ++n) {
        float a = b0[n];
        for (int k = 0; k < 4; ++k) a += x[k] * W0[k * 5 + n];
        pre0[n] = a;
        h[n] = splus(a);
    }
    float preh[4][5];
    for (int L = 0; L < 4; ++L) {
        float hn[5];
        for (int n = 0; n < 5; ++n) {
            float a = bh[L * 5 + n];
            for (int k = 0; k < 5; ++k) a += h[k] * Wh[L * 25 + k * 5 + n];
            preh[L][n] = a;
            hn[n] = splus(a);
        }
        for (int n = 0; n < 5; ++n) h[n] = hn[n];
    }
    float pref = bf[0];
    for (int k = 0; k < 5; ++k) pref += h[k] * Wf[k];
    float e_id = splus(pref);

    // backward
    float g[5];
    float sf = sigmoidf(pref);
    for (int k = 0; k < 5; ++k) g[k] = Wf[k] * sf;
    for (int L = 3; L >= 0; --L) {
        float gn[5];
        for (int k = 0; k < 5; ++k) {
            float a = 0.f;
            for (int n = 0; n < 5; ++n)
                a += Wh[L * 25 + k * 5 + n] * g[n] * sigmoidf(preh[L][n]);
            gn[k] = a;
        }
        for (int k = 0; k < 5; ++k) g[k] = gn[k];
    }
    float gx[4];
    for (int k = 0; k < 4; ++k) {
        float a = 0.f;
        for (int n = 0; n < 5; ++n) a += W0[k * 5 + n] * g[n] * sigmoidf(pre0[n]);
        gx[k] = a;
    }
    float hc = -(2.f * gx[0] + gx[1] - gx[2] + 4.f * gx[3]);
    cst[0] = e_id;
    cst[1] = hc;
}

// --------------------------------------------------------------------------
// Per-lane invariants of one 3x3 F (row-major): trC, det(F), I2
// --------------------------------------------------------------------------
__device__ __forceinline__ void invariants(const float* __restrict__ Fp,
                                           float& trC, float& J, float& I2) {
    float f0 = Fp[0], f1 = Fp[1], f2 = Fp[2];
    float f3 = Fp[3], f4 = Fp[4], f5 = Fp[5];
    float f6 = Fp[6], f7 = Fp[7], f8 = Fp[8];
    // C = F^T F (symmetric)
    float C00 = f0 * f0 + f3 * f3 + f6 * f6;
    float C11 = f1 * f1 + f4 * f4 + f7 * f7;
    float C22 = f2 * f2 + f5 * f5 + f8 * f8;
    float C01 = f0 * f1 + f3 * f4 + f6 * f7;
    float C02 = f0 * f2 + f3 * f5 + f6 * f8;
    float C12 = f1 * f2 + f4 * f5 + f7 * f8;
    trC = C00 + C11 + C22;
    float trC2 = C00 * C00 + C11 * C11 + C22 * C22
               + 2.f * (C01 * C01 + C02 * C02 + C12 * C12);
    I2 = 0.5f * (trC * trC - trC2);
    J = f0 * (f4 * f8 - f5 * f7) - f1 * (f3 * f8 - f5 * f6) + f2 * (f3 * f7 - f4 * f6);
}

// --------------------------------------------------------------------------
// Main kernel: each wave32 handles 32 samples as two 16-column WMMA chains.
// Layout (transposed MLP): D(out_feat x sample) = W^T(A 16x4) * H(B 4x16) + C
//   A: lane m (0-15) -> row m, vgpr0 = K(lane<16?0:2), vgpr1 = K+1
//   B: vgpr0: lanes0-15 = K0 row (cols=samples), lanes16-31 = K2; vgpr1: K1/K3
//   D: vgpr r = row r on lanes 0-15 (rows 8-15 upper half are padded garbage)
// Hidden K=5: K0..3 via WMMA, K=4 rank-1 term folded into the C fragment.
// A-fragments are pre-staged zero-padded in LDS so the hot path has no
// conditional loads (keeps EXEC all-ones, no saveexec chains).
// --------------------------------------------------------------------------
__global__ __launch_bounds__(256) void hyper_elastic_kernel(
    const float* __restrict__ F,
    const float* __restrict__ W0, const float* __restrict__ b0,
    const float* __restrict__ Wh, const float* __restrict__ bh,
    const float* __restrict__ Wf, const float* __restrict__ bf,
    const float* __restrict__ cst,
    float* __restrict__ out, int nsamp)
{
    // ---- stage zero-padded A-fragments: ldsA[layer][vgpr][lane]
    __shared__ float ldsA[5][2][32];
    for (int t = threadIdx.x; t < 320; t += 256) {
        int Lx = t >> 6;            // 0..4   (0 = input layer, 1..4 = hidden)
        int v  = (t >> 5) & 1;      // vgpr within fragment
        int ln = t & 31;
        int mm = ln & 15;
        int kk = ((ln < 16) ? 0 : 2) + v;
        float val = 0.f;
        if (mm < 5) {
            val = (Lx == 0) ? W0[kk * 5 + mm] : Wh[(Lx - 1) * 25 + kk * 5 + mm];
        }
        ldsA[Lx][v][ln] = val;
    }
    __syncthreads();

    const int  lane   = threadIdx.x & 31;
    const int  wave   = blockIdx.x * (blockDim.x >> 5) + (threadIdx.x >> 5);
    const int  base   = wave * 32;
    const int  m      = lane & 15;
    const bool lo     = lane < 16;
    const float loMask = lo ? 1.f : 0.f;       // register-only cndmask

    // ---- invariants for both 16-sample blocks (upper half duplicates hit cache)
    int iA = base + m;        if (iA >= nsamp) iA = nsamp - 1;
    int iB = base + 16 + m;   if (iB >= nsamp) iB = nsamp - 1;
    float trCA, JA, I2A, trCB, JB, I2B;
    invariants(F + (size_t)iA * 9, trCA, JA, I2A);
    invariants(F + (size_t)iB * 9, trCB, JB, I2B);

    // ---- layer 0: x = [trC, J, -J, I2], K = 4 exactly
    v2f a0;
    a0.x = ldsA[0][0][lane];
    a0.y = ldsA[0][1][lane];

    v2f bA, bB;
    bA.x = lo ? trCA : -JA;   bA.y = lo ? JA : I2A;
    bB.x = lo ? trCB : -JB;   bB.y = lo ? JB : I2B;

    v8f c0 = {};
#pragma unroll
    for (int r = 0; r < 5; ++r) {
        float bb = b0[r];                      // uniform s_load
        c0[r] = bb * loMask;
    }

    v8f dA = __builtin_amdgcn_wmma_f32_16x16x4_f32(false, a0, false, bA, (short)0, c0, false, false);
    v8f dB = __builtin_amdgcn_wmma_f32_16x16x4_f32(false, a0, false, bB, (short)0, c0, false, false);

    float sA[5], sB[5];
#pragma unroll
    for (int r = 0; r < 5; ++r) { sA[r] = splus(dA[r]); sB[r] = splus(dB[r]); }

    // ---- 4 hidden layers, 5->5
#pragma unroll
    for (int L = 0; L < 4; ++L) {
        v2f ah;
        ah.x = ldsA[L + 1][0][lane];
        ah.y = ldsA[L + 1][1][lane];

        // B fragments: K2/K3 rows moved into lanes 16-31 via swizzle
        float tA2 = swap16(sA[2]), tA3 = swap16(sA[3]);
        float tB2 = swap16(sB[2]), tB3 = swap16(sB[3]);
        v2f bAn, bBn;
        bAn.x = lo ? sA[0] : tA2;  bAn.y = lo ? sA[1] : tA3;
        bBn.x = lo ? sB[0] : tB2;  bBn.y = lo ? sB[1] : tB3;

        // C fragments: bias (lower rows only) + rank-1 K=4 term (all lanes ok:
        // upper-half rows 8-15 are padded garbage that never propagates)
        v8f ccA = {}, ccB = {};
#pragma unroll
        for (int r = 0; r < 5; ++r) {
            float w4   = Wh[L * 25 + 20 + r];  // uniform s_load
            float bias = bh[L * 5 + r];        // uniform s_load
            float cb   = bias * loMask;
            ccA[r] = fmaf(w4, sA[4], cb);
            ccB[r] = fmaf(w4, sB[4], cb);
        }

        dA = __builtin_amdgcn_wmma_f32_16x16x4_f32(false, ah, false, bAn, (short)0, ccA, false, false);
        dB = __builtin_amdgcn_wmma_f32_16x16x4_f32(false, ah, false, bBn, (short)0, ccB, false, false);
#pragma unroll
        for (int r = 0; r < 5; ++r) { sA[r] = splus(dA[r]); sB[r] = splus(dB[r]); }
    }

    // ---- final 5->1 layer: all features live per-lane -> plain FMAs
    float eA = bf[0], eB = bf[0];
#pragma unroll
    for (int k = 0; k < 5; ++k) {
        float w = Wf[k];                       // uniform s_load
        eA = fmaf(sA[k], w, eA);
        eB = fmaf(sB[k], w, eB);
    }
    eA = splus(eA);
    eB = splus(eB);

    float e_id = cst[0];
    float hc   = cst[1];
    float resA = eA - e_id + hc * 0.5f * (trCA - 3.f);
    float resB = eB - e_id + hc * 0.5f * (trCB - 3.f);

    if (lo) {
        int oA = base + m;
        int oB = base + 16 + m;
        if (oA < nsamp) out[oA] = resA;
        if (oB < nsamp) out[oB] = resB;
    }
}

extern "C" void kernel_launch(void* const* d_in, const int* in_sizes, int n_in,
                              void* d_out, int out_size, void* d_ws, size_t ws_size,
                              hipStream_t stream) {
    const float* F  = (const float*)d_in[0];
    const float* W0 = (const float*)d_in[1];
    const float* b0 = (const float*)d_in[2];
    const float* Wh = (const float*)d_in[3];
    const float* bh = (const float*)d_in[4];
    const float* Wf = (const float*)d_in[5];
    const float* bf = (const float*)d_in[6];
    float* out = (float*)d_out;
    float* cst = (float*)d_ws;   // [e_id, hc]

    int nsamp = in_sizes[0] / 9;

    hyper_setup_kernel<<<1, 1, 0, stream>>>(W0, b0, Wh, bh, Wf, bf, cst);

    int waves  = (nsamp + 31) / 32;
    int blocks = (waves + 7) / 8;    // 256 threads = 8 waves per block
    hyper_elastic_kernel<<<blocks, 256, 0, stream>>>(F, W0, b0, Wh, bh, Wf, bf,
                                                     cst, out, nsamp);
}